// PredictorCore_5720896438559
// MI455X (gfx1250) — compile-verified
//
#include <hip/hip_runtime.h>
#include <stdint.h>

typedef unsigned int u32;
typedef unsigned short u16;
typedef unsigned long long u64;

#define NPRI 400000
#define NCLS 81
#define NCAND 1000
#define CAP 2048
#define SCORE_THR 0.3f
#define IOU_THR 0.5f

typedef __attribute__((ext_vector_type(2))) float v2f;
typedef __attribute__((ext_vector_type(8))) float v8f;
typedef int v4i __attribute__((vector_size(16)));   // matches builtin's V4i param

// ---------- feature detection (compile-only loop: keep fallbacks) ----------
#if defined(__has_builtin)
#if __has_builtin(__builtin_amdgcn_global_load_async_to_lds_b128) && \
    __has_builtin(__builtin_amdgcn_s_wait_asynccnt)
#define USE_ASYNC 1
#endif
#if __has_builtin(__builtin_amdgcn_wmma_f32_16x16x4_f32)
#define USE_WMMA 1
#endif
#endif

#if defined(USE_ASYNC)
#define GAS __attribute__((address_space(1)))
#define LAS __attribute__((address_space(3)))
// AS1 ptr: same bit pattern as flat global addr. AS3 ptr: low-32 truncation of
// flat LDS addr.
__device__ __forceinline__ void async_b128(const void* gsrc, void* ldst) {
  __builtin_amdgcn_global_load_async_to_lds_b128(
      (GAS v4i*)(uintptr_t)gsrc, (LAS v4i*)(u32)(uintptr_t)ldst, 0, 0);
}
#endif

// ---------- orderable float keys ----------
__device__ __forceinline__ u32 flipkey(float f) {
  u32 u = __float_as_uint(f);
  return (u & 0x80000000u) ? ~u : (u | 0x80000000u);
}
__device__ __forceinline__ float unflip(u32 k) {
  u32 u = (k & 0x80000000u) ? (k & 0x7FFFFFFFu) : ~k;
  return __uint_as_float(u);
}

// ---------- ws layout (all offsets 16B aligned) ----------
#define BEST_OFF   0ull        // f32[400000]
#define LABEL_OFF  1600000ull  // i32[400000]
#define HIST_OFF   3200000ull  // u32[256]
#define STATE_OFF  3201024ull  // u32[4] : {prefix, K, cnt, pad}
#define CAND_OFF   3201040ull  // u64[2048]
#define TS_OFF     3217424ull  // f32[1024] top scores
#define TL_OFF     3221520ull  // i32[1024] top labels (0..79)
#define RBOX_OFF   3225616ull  // f32[1024*4] raw boxes
#define OBOX_OFF   3242000ull  // f32[1024*4] offset boxes
#define AREA_OFF   3258384ull  // f32[1024]
#define MASK_OFF   3262480ull  // u16[1024*64] suppression bitmask (bit j%16 of tile j/16)

// ======================= kernel 0: init =======================
__global__ void init_kernel(u32* hist, u32* state) {
  if (threadIdx.x < 256) hist[threadIdx.x] = 0;
  if (threadIdx.x == 0) { state[0] = 0; state[1] = NCAND; state[2] = 0; state[3] = 0; }
}

// ======================= kernel 1: per-prior max/argmax over classes 1..80 =======================
// 128 priors per block; stage 128*81 f32 (41472B, 16B-multiple, 16B-aligned since 1296 | block base)
__global__ void best_class_kernel(const float* __restrict__ scores,
                                  float* __restrict__ best, int* __restrict__ label) {
  __shared__ alignas(16) float sm[128 * NCLS];
  const int bp = blockIdx.x * 128;
  const float* src = scores + (size_t)bp * NCLS;
#if defined(USE_ASYNC)
  for (int c4 = threadIdx.x; c4 < (128 * NCLS) / 4; c4 += 256)
    async_b128(src + c4 * 4, &sm[c4 * 4]);
  __builtin_amdgcn_s_wait_asynccnt(0);
#else
  for (int c = threadIdx.x; c < 128 * NCLS; c += 256) sm[c] = src[c];
#endif
  __syncthreads();
  if (threadIdx.x < 128) {
    const float* row = &sm[(u32)threadIdx.x * NCLS];
    float b = row[1]; int l = 0;
    for (int c = 2; c <= 80; ++c) { float v = row[c]; if (v > b) { b = v; l = c - 1; } }
    int p = bp + threadIdx.x;   // 3125*128 == 400000 exactly
    best[p] = b; label[p] = l;
  }
}

// ======================= kernel 2a: radix-select histogram pass =======================
__global__ void hist_kernel(const float* __restrict__ best, u32* __restrict__ hist,
                            const u32* __restrict__ state, int pass) {
  __shared__ u32 lh[256];
  if (threadIdx.x < 256) lh[threadIdx.x] = 0;
  __syncthreads();
  int p = blockIdx.x * 256 + threadIdx.x;
  if (p < NPRI) {
    u32 key = flipkey(best[p]);
    bool match = true;
    if (pass < 3) { int sh = 8 * (pass + 1); match = (key >> sh) == (state[0] >> sh); }
    if (match) atomicAdd(&lh[(key >> (8 * pass)) & 255u], 1u);
  }
  __syncthreads();
  if (threadIdx.x < 256) { u32 v = lh[threadIdx.x]; if (v) atomicAdd(&hist[threadIdx.x], v); }
}

// ======================= kernel 2b: pick bin, update prefix/K, zero hist =======================
__global__ void scan_kernel(u32* hist, u32* state, int pass) {
  if (threadIdx.x == 0) {
    u32 K = state[1], c = 0, above = 0; int chosen = 0;
    for (int bin = 255; bin >= 0; --bin) {
      u32 h = hist[bin];
      if (c + h >= K) { chosen = bin; above = c; break; }
      c += h;
    }
    state[0] |= ((u32)chosen) << (8 * pass);
    state[1] = K - above;
  }
  __syncthreads();
  if (threadIdx.x < 256) hist[threadIdx.x] = 0;
}

// ======================= kernel 2c: collect keys >= T =======================
__global__ void collect_kernel(const float* __restrict__ best, u32* __restrict__ state,
                               u64* __restrict__ cand) {
  int p = blockIdx.x * 256 + threadIdx.x;
  if (p >= NPRI) return;
  u32 key = flipkey(best[p]);
  if (key >= state[0]) {
    u32 pos = atomicAdd(&state[2], 1u);
    if (pos < CAP) cand[pos] = ((u64)key << 32) | (u32)(~(u32)p);  // key desc, idx asc
  }
}

// ======================= kernel 3: bitonic sort (desc) + gather top-1000 =======================
__global__ void sort_gather_kernel(const u64* __restrict__ cand, const u32* __restrict__ state,
                                   const float* __restrict__ boxes, const int* __restrict__ labels,
                                   float* __restrict__ tscore, int* __restrict__ tlabel,
                                   float* __restrict__ rbox, float* __restrict__ obox,
                                   float* __restrict__ area) {
  __shared__ u64 s[CAP];
  u32 cnt = state[2]; if (cnt > CAP) cnt = CAP;
  for (int e = threadIdx.x; e < CAP; e += 1024) s[e] = (e < (int)cnt) ? cand[e] : 0ull;
  __syncthreads();
  for (int k = 2; k <= CAP; k <<= 1) {
    for (int j = k >> 1; j > 0; j >>= 1) {
      for (int e = threadIdx.x; e < CAP; e += 1024) {
        int ix = e ^ j;
        if (ix > e) {
          u64 x = s[e], y = s[ix];
          bool sw = ((e & k) == 0) ? (x < y) : (x > y);   // overall descending
          if (sw) { s[e] = y; s[ix] = x; }
        }
      }
      __syncthreads();
    }
  }
  int i = threadIdx.x;
  if (i < 1024) {
    if (i < NCAND) {
      u64 c = s[i];
      u32 key = (u32)(c >> 32);
      if (key != 0) {
        u32 idx = ~(u32)c;
        float sc = unflip(key);
        int lb = labels[idx];
        float b0 = boxes[(size_t)idx * 4 + 0], b1 = boxes[(size_t)idx * 4 + 1];
        float b2 = boxes[(size_t)idx * 4 + 2], b3 = boxes[(size_t)idx * 4 + 3];
        float off = (float)lb * 4.0f;
        tscore[i] = sc; tlabel[i] = lb;
        rbox[i * 4 + 0] = b0; rbox[i * 4 + 1] = b1; rbox[i * 4 + 2] = b2; rbox[i * 4 + 3] = b3;
        obox[i * 4 + 0] = b0 + off; obox[i * 4 + 1] = b1 + off;
        obox[i * 4 + 2] = b2 + off; obox[i * 4 + 3] = b3 + off;
        area[i] = (b2 - b0) * (b3 - b1);
        return;
      }
    }
    tscore[i] = 0.f; tlabel[i] = 0; area[i] = 0.f;
    rbox[i * 4 + 0] = rbox[i * 4 + 1] = rbox[i * 4 + 2] = rbox[i * 4 + 3] = 0.f;
    obox[i * 4 + 0] = obox[i * 4 + 1] = obox[i * 4 + 2] = obox[i * 4 + 3] = 0.f;
  }
}

// ======================= kernel 4: IoU tiles -> bitmask (WMMA area outer-sum) =======================
// 63x63 tiles of 16x16 (covers 1008 >= 1000, padded rows are zero boxes/areas).
// D[M][N] = area_i[M] + area_j[N] via V_WMMA_F32_16X16X4_F32 (exact: rank-2, rest zeros).
#define NTILE 63
__global__ void iou_kernel(const float* __restrict__ area, const float* __restrict__ obox,
                           u16* __restrict__ mask16) {
  int wave = threadIdx.x >> 5;
  int lane = threadIdx.x & 31;
  int tile = blockIdx.x * 8 + wave;
  if (tile >= NTILE * NTILE) return;
  int tr = tile / NTILE, tc = tile % NTILE;
  int i0 = tr * 16, j0 = tc * 16;

  v8f acc = {0.f, 0.f, 0.f, 0.f, 0.f, 0.f, 0.f, 0.f};
#if defined(USE_WMMA)
  v2f a = {0.f, 0.f}, b = {0.f, 0.f};
  if (lane < 16) {                 // A: 16x4, lanes0-15 hold K=0,1 ; lanes16-31 K=2,3 (zeros)
    a.x = area[i0 + lane]; a.y = 1.0f;   // A[M][0]=area_i, A[M][1]=1
    b.x = 1.0f; b.y = area[j0 + lane];   // B[0][N]=1,      B[1][N]=area_j
  }
  acc = __builtin_amdgcn_wmma_f32_16x16x4_f32(false, a, false, b, (short)0, acc, false, false);
#else
  {
    int N = lane & 15;
    for (int r = 0; r < 8; ++r) {
      int M = (lane < 16) ? r : (r + 8);
      acc[r] = area[i0 + M] + area[j0 + N];
    }
  }
#endif

  int N = lane & 15;
  float cx1 = obox[(j0 + N) * 4 + 0], cy1 = obox[(j0 + N) * 4 + 1];
  float cx2 = obox[(j0 + N) * 4 + 2], cy2 = obox[(j0 + N) * 4 + 3];
  for (int r = 0; r < 8; ++r) {
    int M = (lane < 16) ? r : (r + 8);
    int row = i0 + M;
    float rx1 = obox[row * 4 + 0], ry1 = obox[row * 4 + 1];
    float rx2 = obox[row * 4 + 2], ry2 = obox[row * 4 + 3];
    float ix = fminf(rx2, cx2) - fmaxf(rx1, cx1);
    float iy = fminf(ry2, cy2) - fmaxf(ry1, cy1);
    float inter = fmaxf(ix, 0.f) * fmaxf(iy, 0.f);
    float uni = acc[r] - inter;                        // area_i + area_j - inter
    bool pred = (inter / fmaxf(uni, 1e-9f)) > IOU_THR;
    u64 bal = __ballot((int)pred);
    if (lane == 0) {                                   // lanes0-15 -> row i0+r ; lanes16-31 -> row i0+r+8
      mask16[(size_t)(i0 + r) * 64 + tc]     = (u16)(bal & 0xFFFFu);
      mask16[(size_t)(i0 + r + 8) * 64 + tc] = (u16)((bal >> 16) & 0xFFFFu);
    }
  }
}

// ======================= kernel 5: sequential NMS (one wave32) + outputs =======================
// keep state: 1024 bits = one u32 per lane. Bitmask (125KB) resident in LDS (CDNA5: 320KB WGP LDS).
__global__ void nms_out_kernel(const u32* __restrict__ gmask, const float* __restrict__ tscore,
                               const int* __restrict__ tlabel, const float* __restrict__ rbox,
                               float* __restrict__ out) {
  __shared__ alignas(16) u32 lmask[32000];   // 1000 rows * 32 words
  __shared__ u32 keepw[32];
  int lane = threadIdx.x;

#if defined(USE_ASYNC)
  for (int c4 = lane; c4 < 32000 / 4; c4 += 32)
    async_b128(gmask + c4 * 4, &lmask[c4 * 4]);
  __builtin_amdgcn_s_wait_asynccnt(0);
#else
  for (int c = lane; c < 32000; c += 32) lmask[c] = gmask[c];
#endif
  __syncthreads();

  u32 kw = 0;
  for (int bnum = 0; bnum < 32; ++bnum) {
    int j = lane * 32 + bnum;
    if (j < NCAND && tscore[j] > SCORE_THR) kw |= (1u << bnum);
  }

  u32 rcur = lmask[lane];                    // row 0 prefetched
  for (int i = 0; i < NCAND; ++i) {
    u32 rnext = (i < NCAND - 1) ? lmask[(i + 1) * 32 + lane] : 0u;
    u32 kword = (u32)__shfl((int)kw, i >> 5);
    if ((kword >> (i & 31)) & 1u) {          // wave-uniform branch
      int wq = i >> 5;
      u32 m;
      if (lane < wq)      m = 0u;
      else if (lane > wq) m = ~0u;
      else                m = ((i & 31) == 31) ? 0u : (~0u << ((i & 31) + 1));  // bits j>i
      kw &= ~(rcur & m);
    }
    rcur = rnext;
  }

  keepw[lane] = kw;
  __syncthreads();

  for (int j = lane; j < NCAND; j += 32) {
    bool k = (keepw[j >> 5] >> (j & 31)) & 1u;
    float kf = k ? 1.0f : 0.0f;
    out[j * 4 + 0] = rbox[j * 4 + 0] * kf;
    out[j * 4 + 1] = rbox[j * 4 + 1] * kf;
    out[j * 4 + 2] = rbox[j * 4 + 2] * kf;
    out[j * 4 + 3] = rbox[j * 4 + 3] * kf;
    out[4000 + j] = k ? (float)(tlabel[j] + 1) : 0.0f;   // sel_labels
    out[5000 + j] = tscore[j] * kf;                      // sel_probs
    out[6000 + j] = kf;                                  // keep
  }
}

// ======================= launch =======================
extern "C" void kernel_launch(void* const* d_in, const int* in_sizes, int n_in,
                              void* d_out, int out_size, void* d_ws, size_t ws_size,
                              hipStream_t stream) {
  (void)in_sizes; (void)n_in; (void)out_size; (void)ws_size;
  const float* scores = (const float*)d_in[0];
  const float* boxes  = (const float*)d_in[1];
  float* out = (float*)d_out;
  char* ws = (char*)d_ws;

  float* best   = (float*)(ws + BEST_OFF);
  int*   label  = (int*)(ws + LABEL_OFF);
  u32*   hist   = (u32*)(ws + HIST_OFF);
  u32*   state  = (u32*)(ws + STATE_OFF);
  u64*   cand   = (u64*)(ws + CAND_OFF);
  float* tscore = (float*)(ws + TS_OFF);
  int*   tlabel = (int*)(ws + TL_OFF);
  float* rbox   = (float*)(ws + RBOX_OFF);
  float* obox   = (float*)(ws + OBOX_OFF);
  float* area   = (float*)(ws + AREA_OFF);
  u16*   mask   = (u16*)(ws + MASK_OFF);

  init_kernel<<<1, 256, 0, stream>>>(hist, state);
  best_class_kernel<<<NPRI / 128, 256, 0, stream>>>(scores, best, label);
  for (int pass = 3; pass >= 0; --pass) {
    hist_kernel<<<(NPRI + 255) / 256, 256, 0, stream>>>(best, hist, state, pass);
    scan_kernel<<<1, 256, 0, stream>>>(hist, state, pass);
  }
  collect_kernel<<<(NPRI + 255) / 256, 256, 0, stream>>>(best, state, cand);
  sort_gather_kernel<<<1, 1024, 0, stream>>>(cand, state, boxes, label,
                                             tscore, tlabel, rbox, obox, area);
  iou_kernel<<<(NTILE * NTILE + 7) / 8, 256, 0, stream>>>(area, obox, mask);
  nms_out_kernel<<<1, 32, 0, stream>>>((const u32*)mask, tscore, tlabel, rbox, out);
}